// GraphAttention2_90039694393674
// MI455X (gfx1250) — compile-verified
//
#include <hip/hip_runtime.h>

typedef __attribute__((ext_vector_type(2))) float        v2f;
typedef __attribute__((ext_vector_type(8))) float        v8f;
typedef __attribute__((ext_vector_type(4))) unsigned int v4u;
typedef __attribute__((ext_vector_type(8))) int          v8i;
typedef __attribute__((ext_vector_type(4))) int          v4i;

namespace {
constexpr int kCin  = 128;
constexpr int kCout = 64;
}

// ---- deg[r] = 1 (self-loop) ------------------------------------------------
__global__ __launch_bounds__(256) void k_deg_init(unsigned* deg, int n) {
  int i = blockIdx.x * 256 + threadIdx.x;
  if (i < n) deg[i] = 1u;
}

// ---- deg[row_e] += 1 for every real edge ----------------------------------
__global__ __launch_bounds__(256) void k_deg_count(const int* __restrict__ ei,
                                                   unsigned* deg, int e) {
  int i = blockIdx.x * 256 + threadIdx.x;
  if (i < e) atomicAdd(&deg[ei[2 * i]], 1u);
}

// ---- S[r] = sum_o (X @ W)[r, o] via V_WMMA_F32_16X16X4_F32 -----------------
// Block = 256 threads = 8 waves; each wave computes a 16x64 tile of H = X@W
// with K=128 (32 wmma steps x 4 column tiles), then row-sums it.
// W is staged in LDS by the Tensor Data Mover (1x8192 f32 tile descriptor,
// wave 0 issues, S_WAIT_TENSORCNT 0 + workgroup barrier completes it).
__global__ __launch_bounds__(256) void k_gemm_rowsum(const float* __restrict__ X,
                                                     const float* __restrict__ W,
                                                     float* __restrict__ S) {
  __shared__ float Wlds[kCin * kCout];                 // 32 KB of 320 KB LDS

#if __has_builtin(__builtin_amdgcn_tensor_load_to_lds) && \
    __has_builtin(__builtin_amdgcn_s_wait_tensorcnt)
  if (threadIdx.x < 32) {                              // one wave issues the DMA
    unsigned           lds_base = (unsigned)(size_t)(&Wlds[0]);
    unsigned long long ga       = (unsigned long long)(size_t)W;
    // D# group 0: count=1 | lds_addr | global_addr[56:0] | type=2
    v4u g0 = { 1u,
               lds_base,
               (unsigned)(ga & 0xffffffffu),
               (unsigned)((ga >> 32) & 0x01ffffffu) | (2u << 30) };
    // D# group 1: data_size=4B; tensor_dim0=tile_dim0=8192; dim1=1; stride=8192
    v8i g1 = { (int)(2u << 16),        // w0: mask=0, data_size=2 (4 bytes)
               (int)(8192u << 16),     // w1: atomic_barrier=0, tensor_dim0.lo
               (int)(1u << 16),        // w2: tensor_dim0.hi=0, tensor_dim1=1
               (int)(8192u << 16),     // w3: tensor_dim1.hi=0, tile_dim0=8192
               1,                      // w4: tile_dim1=1, tile_dim2=0
               8192,                   // w5: tensor_dim0_stride.lo
               0, 0 };                 // w6/w7: stride.hi, dim1_stride
    v4i gz4 = { 0, 0, 0, 0 };          // groups 2/3 unused (<=2-D tensor)
    v8i gz8 = { 0, 0, 0, 0, 0, 0, 0, 0 };
    __builtin_amdgcn_tensor_load_to_lds(g0, g1, gz4, gz4, gz8, 0);
    __builtin_amdgcn_s_wait_tensorcnt(0);              // s_wait_tensorcnt 0
  }
  __syncthreads();
#else
  for (int i = threadIdx.x; i < kCin * kCout; i += 256) Wlds[i] = W[i];
  __syncthreads();
#endif

  const int wave = threadIdx.x >> 5;
  const int lane = threadIdx.x & 31;
  const int half = lane >> 4;       // 0: lanes 0-15, 1: lanes 16-31
  const int m    = lane & 15;
  const int R    = blockIdx.x * 128 + wave * 16;       // 16 rows per wave

  const float* __restrict__ xrow = X + (size_t)(R + m) * kCin + 2 * half;
  __builtin_prefetch(xrow + 64, 0, 0);                 // global_prefetch_b8

  v8f acc0{}, acc1{}, acc2{}, acc3{};
  #pragma unroll 4
  for (int k0 = 0; k0 < kCin; k0 += 4) {
    v2f a = *(const v2f*)(xrow + k0);                  // global_load_b64
    const float* w0 = &Wlds[(k0 + 2 * half) * kCout + m];
    const float* w1 = w0 + kCout;
    v2f b0 = { w0[0],  w1[0]  };
    v2f b1 = { w0[16], w1[16] };
    v2f b2 = { w0[32], w1[32] };
    v2f b3 = { w0[48], w1[48] };
    acc0 = __builtin_amdgcn_wmma_f32_16x16x4_f32(false, a, false, b0, (short)0, acc0, false, false);
    acc1 = __builtin_amdgcn_wmma_f32_16x16x4_f32(false, a, false, b1, (short)0, acc1, false, false);
    acc2 = __builtin_amdgcn_wmma_f32_16x16x4_f32(false, a, false, b2, (short)0, acc2, false, false);
    acc3 = __builtin_amdgcn_wmma_f32_16x16x4_f32(false, a, false, b3, (short)0, acc3, false, false);
  }

  // C/D layout: VGPR i holds row M = i + 8*half, col N = (lane&15) + 16*t.
  // Row-sum: add the 4 column tiles, then reduce across the 16 lanes of each
  // half-wave (xor masks 1,2,4,8 never cross the half boundary).
  #pragma unroll
  for (int i = 0; i < 8; ++i) {
    float v = acc0[i] + acc1[i] + acc2[i] + acc3[i];
    v += __shfl_xor(v, 1);
    v += __shfl_xor(v, 2);
    v += __shfl_xor(v, 4);
    v += __shfl_xor(v, 8);
    if (m == 0) S[R + 8 * half + i] = v;
  }
}

// ---- s-prefix: s[i] for the first N edge slots (only ones ever gathered) ---
__global__ __launch_bounds__(256) void k_sprefix(const int* __restrict__ ei,
                                                 const float* __restrict__ ed,
                                                 const float* __restrict__ S,
                                                 const unsigned* __restrict__ deg,
                                                 float* __restrict__ sp, int n) {
  int i = blockIdx.x * 256 + threadIdx.x;
  if (i >= n) return;
  int r   = ei[2 * i];
  float z = (float)deg[r] + 1e-16f;                    // softmax Z = deg
  sp[i] = 2.0f * S[r] / (z * ed[i]);
}

// ---- d_j = s[row_j] - s[col_j] --------------------------------------------
__global__ __launch_bounds__(256) void k_delta(const int* __restrict__ ei,
                                               const float* __restrict__ sp,
                                               float* __restrict__ out,
                                               int e, int total) {
  int j = blockIdx.x * 256 + threadIdx.x;
  if (j >= total) return;
  int rj, cj;
  if (j < e) { rj = ei[2 * j]; cj = ei[2 * j + 1]; }
  else       { rj = cj = j - e; }                      // self-loop rows: 0
  out[j] = sp[rj] - sp[cj];
}

extern "C" void kernel_launch(void* const* d_in, const int* in_sizes, int n_in,
                              void* d_out, int out_size, void* d_ws, size_t ws_size,
                              hipStream_t stream) {
  const float* x  = (const float*)d_in[0];  // (N, 128)
  const int*   ei = (const int*)d_in[1];    // (E, 2)
  const float* ed = (const float*)d_in[2];  // (E,)
  const float* w  = (const float*)d_in[3];  // (1, 128, 64)
  (void)d_in[4];  // attention vector cancels analytically (uniform softmax)

  const int n     = in_sizes[0] / kCin;     // 65536
  const int e     = in_sizes[2];            // 1048576
  const int total = e + n;                  // == out_size

  unsigned* deg = (unsigned*)d_ws;                               // N u32
  float*    S   = (float*)((char*)d_ws + (size_t)n * 4);         // N f32
  float*    sp  = (float*)((char*)d_ws + (size_t)n * 8);         // N f32
  float*    out = (float*)d_out;

  k_deg_init  <<<(n + 255) / 256, 256, 0, stream>>>(deg, n);
  k_deg_count <<<(e + 255) / 256, 256, 0, stream>>>(ei, deg, e);
  k_gemm_rowsum<<<n / 128,        256, 0, stream>>>(x, w, S);
  k_sprefix   <<<(n + 255) / 256, 256, 0, stream>>>(ei, ed, S, deg, sp, n);
  k_delta     <<<(total + 255) / 256, 256, 0, stream>>>(ei, sp, out, e, total);

  (void)n_in; (void)out_size; (void)ws_size;
}